// CubeNet_87093346828907
// MI455X (gfx1250) — compile-verified
//
#include <hip/hip_runtime.h>
#include <hip/hip_bf16.h>

typedef __attribute__((ext_vector_type(2))) float v2f;
typedef __attribute__((ext_vector_type(8))) float v8f;

#define N_PTS 384
#define DD    128
#define EPSF  1e-6f

// ---------------- WMMA helpers (V_WMMA_F32_16X16X4_F32, wave32) ----------------

__device__ __forceinline__ v8f wmma_f32(v2f a, v2f b, v8f c) {
  // 8 args: (neg_a, A, neg_b, B, c_mod, C, reuse_a, reuse_b)
  return __builtin_amdgcn_wmma_f32_16x16x4_f32(false, a, false, b, (short)0, c,
                                               false, false);
}

// A-fragment (16x4 MxK) from row-major S: elem[m,k] = S[(row0+m)*ld + k0+k]
// Lanes 0-15: K = k0,k0+1 ; lanes 16-31: K = k0+2,k0+3 (ISA 7.12.2)
__device__ __forceinline__ v2f frag_rm(const float* __restrict__ S, int ld,
                                       int row0, int k0, int lane) {
  int m  = row0 + (lane & 15);
  int kk = k0 + ((lane >> 4) << 1);
  v2f f;
  f.x = S[m * ld + kk];
  f.y = S[m * ld + kk + 1];
  return f;
}

// B-fragment (4x16 KxN) from row-major S: elem[k,n] = S[(k0+k)*ld + col0+n]
__device__ __forceinline__ v2f frag_cm(const float* __restrict__ S, int ld,
                                       int k0, int col0, int lane) {
  int n  = col0 + (lane & 15);
  int kk = k0 + ((lane >> 4) << 1);
  v2f f;
  f.x = S[kk * ld + n];
  f.y = S[(kk + 1) * ld + n];
  return f;
}

// C/D 16x16 f32: VGPR r -> row r (lanes 0-15, N=lane) / row r+8 (lanes 16-31)
__device__ __forceinline__ void store_c(float* __restrict__ C, int ldc,
                                        int row0, int col0, int lane, v8f c) {
  int col = col0 + (lane & 15);
  int rb  = row0 + ((lane >> 4) << 3);
#pragma unroll
  for (int r = 0; r < 8; ++r) C[(rb + r) * ldc + col] = c[r];
}

// ---------------- GEMM kernels (one wave per 16x16 output tile) ----------------

// Out[128x128] = In^T @ In   (In is 128x128 row-major)
__global__ void k_ata(const float* __restrict__ In, float* __restrict__ Out) {
  int lane = threadIdx.x;
  int row0 = blockIdx.y * 16, col0 = blockIdx.x * 16;
  v8f c = {};
#pragma unroll 4
  for (int k0 = 0; k0 < DD; k0 += 4) {
    v2f a = frag_cm(In, DD, k0, row0, lane);  // A[m,k] = In[k, row0+m]
    v2f b = frag_cm(In, DD, k0, col0, lane);  // B[k,n] = In[k, col0+n]
    c = wmma_f32(a, b, c);
  }
  store_c(Out, DD, row0, col0, lane, c);
}

// Aout[z] = X[384x128] @ Wz[128x128], z in {0,1}
__global__ void k_xw(const float* __restrict__ X, const float* __restrict__ W0,
                     const float* __restrict__ W1, float* __restrict__ Aout) {
  int lane = threadIdx.x;
  int row0 = blockIdx.y * 16, col0 = blockIdx.x * 16;
  const float* W = (blockIdx.z == 0) ? W0 : W1;
  float* Out = Aout + (size_t)blockIdx.z * (N_PTS * DD);
  v8f c = {};
#pragma unroll 4
  for (int k0 = 0; k0 < DD; k0 += 4) {
    v2f a = frag_rm(X, DD, row0, k0, lane);
    v2f b = frag_cm(W, DD, k0, col0, lane);
    c = wmma_f32(a, b, c);
  }
  store_c(Out, DD, row0, col0, lane, c);
}

// Cross[z] = Ain[z][384x128] @ X[384x128]^T
__global__ void k_abt(const float* __restrict__ Ain, const float* __restrict__ X,
                      float* __restrict__ Cross) {
  int lane = threadIdx.x;
  int row0 = blockIdx.y * 16, col0 = blockIdx.x * 16;
  const float* A = Ain + (size_t)blockIdx.z * (N_PTS * DD);
  float* Out = Cross + (size_t)blockIdx.z * (N_PTS * N_PTS);
  v8f c = {};
#pragma unroll 4
  for (int k0 = 0; k0 < DD; k0 += 4) {
    v2f a = frag_rm(A, DD, row0, k0, lane);
    v2f b = frag_rm(X, DD, col0, k0, lane);  // B[k,n] = X[col0+n, k]
    c = wmma_f32(a, b, c);
  }
  store_c(Out, N_PTS, row0, col0, lane, c);
}

// ---------------- Elementwise / reduction kernels ----------------

__global__ void k_diag(const float* __restrict__ Cross, float* __restrict__ Diag) {
  int t = threadIdx.x;
  if (t < N_PTS) {
    Diag[t]         = Cross[t * (N_PTS + 1)];
    Diag[N_PTS + t] = Cross[N_PTS * N_PTS + t * (N_PTS + 1)];
  }
}

// In-place: Cross[0] -> csmd ; Cross[1] -> r = 1/max(2*sqrt(max(csmd2,eps)),eps)
__global__ void k_prep(float* __restrict__ Cross, const float* __restrict__ Diag) {
  int idx = blockIdx.x * blockDim.x + threadIdx.x;  // 0 .. 384*384-1
  int i = idx / N_PTS;
  int k = idx - i * N_PTS;
  float c1 = Cross[idx];
  Cross[idx] = Diag[i] + Diag[k] - 2.0f * c1;
  float c2  = Cross[N_PTS * N_PTS + idx];
  float cs2 = Diag[N_PTS + i] + Diag[N_PTS + k] - 2.0f * c2;
  float den = fmaxf(2.0f * sqrtf(fmaxf(cs2, EPSF)), EPSF);
  Cross[N_PTS * N_PTS + idx] = 1.0f / den;
}

// out[i] = min_{k: y[k]!=y[i]} max_{j: y[j]==y[i], j!=i} relu(csmd[i,k]-csmd[i,j])*r[j,k]
__global__ void k_reduce(const float* __restrict__ csmd, const float* __restrict__ r,
                         const int* __restrict__ y, float* __restrict__ out) {
  __shared__ float s_c[N_PTS];
  __shared__ int   s_y[N_PTS];
  __shared__ int   s_valid[N_PTS];
  __shared__ int   s_nvalid;
  __shared__ float s_min[256];

  int i   = blockIdx.x;
  int tid = threadIdx.x;  // 256 threads

  for (int t = tid; t < N_PTS; t += 256) {
    s_c[t] = csmd[i * N_PTS + t];
    s_y[t] = y[t];
  }
  __syncthreads();

  int yi = s_y[i];
  if (tid == 0) {
    int c = 0;
    for (int j = 0; j < N_PTS; ++j)
      if (s_y[j] == yi && j != i) s_valid[c++] = j;
    s_nvalid = c;
  }
  __syncthreads();
  int nv = s_nvalid;

  float best = __builtin_inff();
  for (int k = tid; k < N_PTS; k += 256) {
    if (s_y[k] == yi) continue;  // inner_max forced to +inf -> never the min
    float cik = s_c[k];
    float acc = 0.0f;            // zeroed margins floor the max at 0
    for (int jj = 0; jj < nv; ++jj) {
      int j   = s_valid[jj];
      float d = fmaxf(cik - s_c[j], 0.0f);
      acc = fmaxf(acc, d * r[j * N_PTS + k]);  // coalesced over k
    }
    best = fminf(best, acc);
  }

  s_min[tid] = best;
  __syncthreads();
#pragma unroll
  for (int off = 128; off > 0; off >>= 1) {
    if (tid < off) s_min[tid] = fminf(s_min[tid], s_min[tid + off]);
    __syncthreads();
  }
  if (tid == 0) out[i] = s_min[0];
}

// ---------------- Launcher ----------------

extern "C" void kernel_launch(void* const* d_in, const int* in_sizes, int n_in,
                              void* d_out, int out_size, void* d_ws, size_t ws_size,
                              hipStream_t stream) {
  const float* X = (const float*)d_in[0];  // [384,128]
  const float* G = (const float*)d_in[1];  // [128,128]
  const int*   y = (const int*)d_in[2];    // [384]
  float* out = (float*)d_out;              // [384]

  float* ws    = (float*)d_ws;
  float* M     = ws;                           // 128*128
  float* M2    = M + DD * DD;                  // 128*128
  float* Aout  = M2 + DD * DD;                 // 2 * 384*128
  float* Cross = Aout + 2 * N_PTS * DD;        // 2 * 384*384 (-> csmd, r in place)
  float* Diag  = Cross + 2 * N_PTS * N_PTS;    // 2 * 384

  dim3 w32(32);
  k_ata<<<dim3(8, 8), w32, 0, stream>>>(G, M);            // M = G^T G
  k_ata<<<dim3(8, 8), w32, 0, stream>>>(M, M2);           // M2 = M^T M
  k_xw <<<dim3(8, 24, 2), w32, 0, stream>>>(X, M, M2, Aout);
  k_abt<<<dim3(24, 24, 2), w32, 0, stream>>>(Aout, X, Cross);
  k_diag<<<1, 384, 0, stream>>>(Cross, Diag);
  k_prep<<<(N_PTS * N_PTS) / 256, 256, 0, stream>>>(Cross, Diag);
  k_reduce<<<N_PTS, 256, 0, stream>>>(Cross, Cross + N_PTS * N_PTS, y, out);
}